// GNNLayer_5772436045872
// MI455X (gfx1250) — compile-verified
//
#include <hip/hip_runtime.h>
#include <hip/hip_bf16.h>

// ---------------------------------------------------------------------------
// GCNConv (sym-norm, self loops) + bias + BatchNorm(batch stats) + ReLU
// for MI455X / gfx1250 (wave32, WMMA).
//   x:[N,128] f32, edge_index:[2,E] i64, W:[128,128] f32, b,gamma,beta:[128]
// ---------------------------------------------------------------------------

#define DF 128  // feature dim (both in and out)

typedef float v2f __attribute__((ext_vector_type(2)));
typedef float v8f __attribute__((ext_vector_type(8)));

__device__ __forceinline__ void atomic_add_f32(float* p, float v) {
    __hip_atomic_fetch_add(p, v, __ATOMIC_RELAXED, __HIP_MEMORY_SCOPE_AGENT);
}

// ---- 1) init: deg=1.0 (self loop), zero column stats ----------------------
__global__ void k_init(float* __restrict__ deg, float* __restrict__ stats,
                       int n) {
    int gid = blockIdx.x * blockDim.x + threadIdx.x;
    if (gid < n) deg[gid] = 1.0f;
    if (gid < 2 * DF) stats[gid] = 0.0f;   // colsum + colsumsq
}

// ---- 2) degree accumulation over edge destinations ------------------------
__global__ void k_degree(const long long* __restrict__ ei, float* __restrict__ deg,
                         int e) {
    int gid = blockIdx.x * blockDim.x + threadIdx.x;
    if (gid >= e) return;
    long long d = ei[(size_t)e + gid];     // dst row of edge_index
    atomic_add_f32(&deg[d], 1.0f);
}

// ---- 3) deg -> rsqrt(deg) in place ----------------------------------------
__global__ void k_rsqrt(float* __restrict__ deg, int n) {
    int gid = blockIdx.x * blockDim.x + threadIdx.x;
    if (gid < n) deg[gid] = __frsqrt_rn(deg[gid]);
}

// ---- 4) xw = x @ W via V_WMMA_F32_16X16X4_F32 -----------------------------
// One wave per 16-row tile of x; the wave computes ALL 8 column tiles
// (8 accumulators), so each A fragment is loaded once and reused 8x.
// K=128 -> 32 k-steps, 8 WMMAs each = 256 WMMAs per wave.
__global__ void k_gemm_wmma(const float* __restrict__ x,
                            const float* __restrict__ W,
                            float* __restrict__ xw, int nrows) {
    int tm   = (blockIdx.x * blockDim.x + threadIdx.x) >> 5;  // row tile
    int lane = threadIdx.x & 31;
    if (tm * 16 >= nrows) return;
    int half = lane >> 4;         // 0: lanes 0-15, 1: lanes 16-31
    int m    = lane & 15;

    const float* xrow = x + (size_t)(tm * 16 + m) * DF;

    v8f c[8];
#pragma unroll
    for (int t = 0; t < 8; ++t) c[t] = (v8f){0.f,0.f,0.f,0.f,0.f,0.f,0.f,0.f};

    for (int ks = 0; ks < 32; ++ks) {
        int k0 = ks * 4 + 2 * half;
        // A fragment: x[row][k0], x[row][k0+1]  (8B-aligned float2)
        v2f a = *(const v2f*)(xrow + k0);
        const float* wr0 = W + (size_t)k0 * DF + m;   // row k0
        const float* wr1 = wr0 + DF;                  // row k0+1
#pragma unroll
        for (int t = 0; t < 8; ++t) {
            // B fragment for column tile t: W[k0][t*16+m], W[k0+1][t*16+m]
            v2f b;
            b.x = wr0[t * 16];
            b.y = wr1[t * 16];
            // (neg_a, A, neg_b, B, c_mod, C, reuse_a, reuse_b)
            c[t] = __builtin_amdgcn_wmma_f32_16x16x4_f32(false, a, false, b,
                                                         (short)0, c[t],
                                                         false, false);
        }
    }

    // D layout: VGPR i -> row = tm*16 + i + 8*half, col = t*16 + (lane&15)
    float* obase = xw + (size_t)(tm * 16 + half * 8) * DF + m;
#pragma unroll
    for (int t = 0; t < 8; ++t) {
#pragma unroll
        for (int i = 0; i < 8; ++i) {
            obase[(size_t)i * DF + t * 16] = c[t][i];
        }
    }
}

// ---- 5) out = b + xw * dinv^2  (bias + self-loop message) -----------------
__global__ void k_init_out(const float* __restrict__ xw,
                           const float* __restrict__ dinv,
                           const float* __restrict__ bias,
                           float* __restrict__ out, int n) {
    int gid = blockIdx.x * blockDim.x + threadIdx.x;
    if (gid >= n * DF) return;
    int row = gid >> 7;
    int col = gid & (DF - 1);
    float di = dinv[row];
    out[gid] = bias[col] + xw[gid] * di * di;
}

// ---- 6) edge scatter: one wave per edge, float4 per lane ------------------
__global__ void k_edge_scatter(const long long* __restrict__ ei,
                               const float* __restrict__ xw,
                               const float* __restrict__ dinv,
                               float* __restrict__ out, int e) {
    int wave = (blockIdx.x * blockDim.x + threadIdx.x) >> 5;
    int lane = threadIdx.x & 31;
    if (wave >= e) return;
    long long s = ei[wave];
    long long d = ei[(size_t)e + wave];
    float norm = dinv[s] * dinv[d];
    const float4 v = *(const float4*)(xw + (size_t)s * DF + lane * 4);
    float* dp = out + (size_t)d * DF + lane * 4;
    atomic_add_f32(dp + 0, v.x * norm);
    atomic_add_f32(dp + 1, v.y * norm);
    atomic_add_f32(dp + 2, v.z * norm);
    atomic_add_f32(dp + 3, v.w * norm);
}

// ---- 7) per-column sum / sumsq (coalesced: 128 threads = one row slice) ---
__global__ void k_col_stats(const float* __restrict__ out,
                            float* __restrict__ colsum,
                            float* __restrict__ colsumsq, int n) {
    int col = threadIdx.x;                  // blockDim.x == 128
    float s = 0.f, s2 = 0.f;
    for (int row = blockIdx.x; row < n; row += gridDim.x) {
        float v = out[(size_t)row * DF + col];
        s += v;
        s2 += v * v;
    }
    atomic_add_f32(&colsum[col], s);
    atomic_add_f32(&colsumsq[col], s2);
}

// ---- 8) finalize: scale = gamma*rsqrt(var+eps), shift = beta - mean*scale -
__global__ void k_bn_finalize(const float* __restrict__ colsum,
                              const float* __restrict__ colsumsq,
                              const float* __restrict__ gamma,
                              const float* __restrict__ beta,
                              float* __restrict__ scale,
                              float* __restrict__ shift, int n) {
    int c = threadIdx.x;                    // 128 threads
    float invn = 1.0f / (float)n;
    float mean = colsum[c] * invn;
    float var  = fmaxf(colsumsq[c] * invn - mean * mean, 0.0f);
    float sc = gamma[c] * __frsqrt_rn(var + 1e-5f);
    scale[c] = sc;
    shift[c] = beta[c] - mean * sc;
}

// ---- 9) apply BN + ReLU in place ------------------------------------------
__global__ void k_bn_relu(float* __restrict__ out,
                          const float* __restrict__ scale,
                          const float* __restrict__ shift, int n) {
    int gid = blockIdx.x * blockDim.x + threadIdx.x;
    if (gid >= n * DF) return;
    int col = gid & (DF - 1);
    out[gid] = fmaxf(fmaf(out[gid], scale[col], shift[col]), 0.0f);
}

// ---------------------------------------------------------------------------
extern "C" void kernel_launch(void* const* d_in, const int* in_sizes, int n_in,
                              void* d_out, int out_size, void* d_ws, size_t ws_size,
                              hipStream_t stream) {
    const float*     x     = (const float*)d_in[0];      // [N,128]
    const long long* ei    = (const long long*)d_in[1];  // [2,E] int64
    const float*     W     = (const float*)d_in[2];      // [128,128]
    const float*     bias  = (const float*)d_in[3];      // [128]
    const float*     gamma = (const float*)d_in[4];      // [128]
    const float*     beta  = (const float*)d_in[5];      // [128]
    float*           out   = (float*)d_out;              // [N,128]

    const int N = in_sizes[0] / DF;
    const int E = in_sizes[1] / 2;

    // workspace layout
    float* xw       = (float*)d_ws;            // N*128
    float* dinv     = xw + (size_t)N * DF;     // N   (holds deg, then rsqrt)
    float* colsum   = dinv + N;                // 128
    float* colsumsq = colsum + DF;             // 128
    float* scale    = colsumsq + DF;           // 128
    float* shift    = scale + DF;              // 128

    const int BT = 256;

    // 1) deg = 1 (self loops), zero stats
    k_init<<<(N + BT - 1) / BT, BT, 0, stream>>>(dinv, colsum, N);
    // 2) deg += 1 per incoming edge
    k_degree<<<(E + BT - 1) / BT, BT, 0, stream>>>(ei, dinv, E);
    // 3) dinv = rsqrt(deg)
    k_rsqrt<<<(N + BT - 1) / BT, BT, 0, stream>>>(dinv, N);
    // 4) xw = x @ W  (WMMA): one wave per 16-row tile, 8 col tiles per wave
    {
        int tiles_m = (N + 15) / 16;            // 6250
        int blocks  = (tiles_m + 7) / 8;        // 8 waves per 256-thread block
        k_gemm_wmma<<<blocks, BT, 0, stream>>>(x, W, xw, N);
    }
    // 5) out = b + self-loop term
    k_init_out<<<((size_t)N * DF + BT - 1) / BT, BT, 0, stream>>>(xw, dinv, bias, out, N);
    // 6) scatter messages: one wave per edge
    k_edge_scatter<<<(E + 7) / 8, BT, 0, stream>>>(ei, xw, dinv, out, E);
    // 7) column stats
    k_col_stats<<<1024, DF, 0, stream>>>(out, colsum, colsumsq, N);
    // 8) finalize BN params
    k_bn_finalize<<<1, DF, 0, stream>>>(colsum, colsumsq, gamma, beta, scale, shift, N);
    // 9) BN + ReLU
    k_bn_relu<<<((size_t)N * DF + BT - 1) / BT, BT, 0, stream>>>(out, scale, shift, N);
}